// AngularSymmetry_46961172414989
// MI455X (gfx1250) — compile-verified
//
#include <hip/hip_runtime.h>
#include <hip/hip_bf16.h>

// B=16, N=128. One workgroup per (b,i): 2048 workgroups x 256 threads (8 waves).
// Wave w computes theta row-block jt=w via V_WMMA_F32_16X16X4_F32 (Gram matrix,
// K=3 padded to 4), then the transcendental reduction over its 8 column tiles.
// Per-element math uses single-instruction hardware ops: v_rcp_f32, v_cos_f32,
// v_log_f32, v_exp_f32 (no IEEE div expansion, no libm range reduction).

typedef __attribute__((ext_vector_type(2))) float v2f;
typedef __attribute__((ext_vector_type(8))) float v8f;

#define NN 128
#define EPS_ 1e-5f
#define ZETA_ 0.8f
#define LOG2E_ 1.4426950408889634f
#define INV2PI_ 0.15915494309189535f   /* v_cos_f32 input is in revolutions */
#define SCALE_ 1.1486983549970350f     /* 2^(1-zeta) */

__global__ __launch_bounds__(256) void angsym_kernel(const float* __restrict__ dc,
                                                     const float* __restrict__ d,
                                                     const float* __restrict__ coord,
                                                     float* __restrict__ out) {
    __shared__ float4 s_vec[NN];   // vec_j = coord_i - coord_j, w-component = 0 (K pad)
    __shared__ float  s_d[NN];     // d[b,i,:]
    __shared__ float  s_dc[NN];    // d_cutoff[b,i,:]
    __shared__ float  s_part[8];   // per-wave partial sums

    const int bi = blockIdx.x;          // b*128 + i
    const int b  = bi >> 7;
    const int i  = bi & 127;
    const int tid = threadIdx.x;

    // ---- stage per-(b,i) data into LDS ----
    if (tid < NN) {
        const int j = tid;
        const float* cb = coord + (size_t)b * NN * 3;
        const float ix = cb[i * 3 + 0];
        const float iy = cb[i * 3 + 1];
        const float iz = cb[i * 3 + 2];
        float4 v;
        v.x = ix - cb[j * 3 + 0];
        v.y = iy - cb[j * 3 + 1];
        v.z = iz - cb[j * 3 + 2];
        v.w = 0.0f;                      // pad K=3 -> 4 for WMMA
        s_vec[j] = v;
        const size_t rowbi = ((size_t)b * NN + i) * NN;
        s_d[j]  = d[rowbi + j];
        s_dc[j] = dc[rowbi + j];
    }
    __syncthreads();

    const int wave = tid >> 5;           // 0..7 -> row block jt
    const int lane = tid & 31;
    const int idx  = lane & 15;          // M (for A) / N (for B) within tile
    const int half = lane >> 4;          // selects K = {0,1} or {2,3}

    // A operand (rows j of V): lane m holds V[m][2h], V[m][2h+1] in 2 VGPRs
    const int jt = wave;
    const float4 vj = s_vec[jt * 16 + idx];
    v2f a;
    a.x = half ? vj.z : vj.x;
    a.y = half ? vj.w : vj.y;

    const size_t base_b = (size_t)b * NN * NN;
    const int jbase = jt * 16 + 8 * half; // C layout: lanes 16-31 hold M=8..15
    float acc = 0.0f;

    for (int kt = 0; kt < 8; ++kt) {
        // B operand (cols k of V^T) — same per-lane layout as A for the Gram case
        const float4 vk = s_vec[kt * 16 + idx];
        v2f bm;
        bm.x = half ? vk.z : vk.x;
        bm.y = half ? vk.w : vk.y;

        v8f c = {};
        // theta tile = A(16x4) x B(4x16): emits v_wmma_f32_16x16x4_f32
        c = __builtin_amdgcn_wmma_f32_16x16x4_f32(
                /*neg_a=*/false, a, /*neg_b=*/false, bm,
                /*c_mod=*/(short)0, c, /*reuse_a=*/false, /*reuse_b=*/false);

        const int   k    = kt * 16 + idx;   // fixed per lane in this tile
        const float dik  = s_d[k];
        const float dcik = s_dc[k];
        const float dik2 = dik * dik;
        const float* djk_row  = d  + base_b + k;
        const float* dcjk_row = dc + base_b + k;

#pragma unroll
        for (int v = 0; v < 8; ++v) {
            const int   j    = jbase + v;
            const float dij  = s_d[j];
            const float dcij = s_dc[j];
            const float djk  = djk_row[j * NN];   // d[b,j,k] (L2-resident)
            const float dcjk = dcjk_row[j * NN];  // dc[b,j,k]

            // theta = dot / (R_ij*R_ik + eps): single v_rcp_f32 + v_mul
            const float theta = c[v] * __builtin_amdgcn_rcpf(fmaf(dij, dik, EPS_));
            // cos via hardware v_cos_f32 (argument in revolutions)
            const float x  = fmaxf(1.0f + __builtin_amdgcn_cosf(theta * INV2PI_), 0.0f);
            const float s2 = fmaf(dij, dij, fmaf(djk, djk, dik2));  // R²sum (ITA=1)
            // (1+cosθ)^ζ · exp(-s2) = exp2(ζ·log2 x − log2e·s2); x=0 -> -inf -> 0
            const float e  = __builtin_amdgcn_exp2f(
                fmaf(ZETA_, __builtin_amdgcn_logf(x), -LOG2E_ * s2));
            acc = fmaf(e, dcij * dcik * dcjk, acc);
        }
    }

    // ---- deterministic reduction: shuffle within wave, serial across waves ----
#pragma unroll
    for (int off = 16; off > 0; off >>= 1)
        acc += __shfl_xor(acc, off, 32);
    if (lane == 0) s_part[wave] = acc;
    __syncthreads();
    if (tid == 0) {
        float t = 0.0f;
#pragma unroll
        for (int w = 0; w < 8; ++w) t += s_part[w];
        out[bi] = t * SCALE_;
    }
}

extern "C" void kernel_launch(void* const* d_in, const int* in_sizes, int n_in,
                              void* d_out, int out_size, void* d_ws, size_t ws_size,
                              hipStream_t stream) {
    (void)in_sizes; (void)n_in; (void)d_ws; (void)ws_size; (void)out_size;
    const float* dc    = (const float*)d_in[0]; // d_cutoff (16,128,128)
    const float* d     = (const float*)d_in[1]; // d        (16,128,128)
    const float* coord = (const float*)d_in[2]; // atom_coordinates (16,128,3)
    float* out = (float*)d_out;                 // (16,128)
    angsym_kernel<<<dim3(16 * 128), dim3(256), 0, stream>>>(dc, d, coord, out);
}